// FeynNet_22849226015102
// MI455X (gfx1250) — compile-verified
//
#include <hip/hip_runtime.h>

typedef float v2f __attribute__((ext_vector_type(2)));
typedef float v8f __attribute__((ext_vector_type(8)));

#define BB 128
#define CC 32
#define NN 128
#define PP 8128
#define NBP 1040384.0f   // B*P
#define EPSV 1e-5f
#define NEG_SLOPE 0.01f

#define CHUNK 128        // p elements staged in LDS per round
#define NCHUNK 64        // ceil(PP / CHUNK)
#define SPLIT 8          // p-splits for the stats pass (occupancy)
#define CHUNKS_PER (NCHUNK / SPLIT)
#define NSLOT 4          // replicated accumulator slots (atomic contention /4)
#define WSG 1056         // per-slot stride: 1024 Gram + 32 Sum
#define HS 130           // LDS h row stride (even -> 8B-aligned b64 pairs)

__global__ void zero_kernel(float* __restrict__ p, int n) {
  int i = blockIdx.x * blockDim.x + threadIdx.x;
  if (i < n) p[i] = 0.0f;
}

// ---------------- Pass 1: per-channel sums + 32x32 Gram of h ----------------
__global__ __launch_bounds__(256)
void stats_kernel(const float* __restrict__ bsrc, const int* __restrict__ assign,
                  float* __restrict__ acc_ws)
{
  __shared__ float brow[CC * NN];      // 16 KB: b[batch] tile
  __shared__ float h[CC][HS];          // h chunk
  __shared__ float gsh[CC * CC];       // block-local Gram
  __shared__ float sred[256];

  const int t = threadIdx.x;
  const int lane = t & 31;
  const int wave = t >> 5;
  const int batch = blockIdx.x;
  const int chunk0 = blockIdx.y * CHUNKS_PER;

  const float* bp = bsrc + batch * (CC * NN);
  for (int i = t; i < CC * NN; i += 256) brow[i] = bp[i];
  for (int i = t; i < CC * CC; i += 256) gsh[i] = 0.0f;

  v8f c00 = {}, c01 = {}, c10 = {}, c11 = {};
  float spart = 0.0f;
  const int sc = t & 31;       // channel for sum reduction
  const int sj = t >> 5;       // p-phase for sum reduction

  for (int chunk = chunk0; chunk < chunk0 + CHUNKS_PER; ++chunk) {
    const int pbase = chunk * CHUNK;
    __syncthreads();
    // recompute h[c][pl] for this chunk (zero-fill OOB p -> no-op for sums/Gram)
    #pragma unroll 4
    for (int i = 0; i < 16; ++i) {
      int f = t + 256 * i;
      int c = f >> 7;
      int pl = f & 127;
      int p = pbase + pl;
      float v = 0.0f;
      if (p < PP) {
        int a0 = assign[2 * p];
        int a1 = assign[2 * p + 1];
        v = fmaxf(brow[c * NN + a0], brow[c * NN + a1]);
      }
      h[c][pl] = v;
    }
    __syncthreads();

    // per-channel sums
    #pragma unroll
    for (int i = 0; i < 16; ++i) spart += h[sc][sj + 8 * i];

    // Gram via f32 WMMA: wave owns p strip [wave*16, wave*16+16)
    const int pl0 = wave * 16;
    #pragma unroll
    for (int s = 0; s < 4; ++s) {
      const int pb = pl0 + 4 * s;
      v2f hlo, hhi;
      if (lane < 16) {
        hlo.x = h[lane][pb + 0];      hlo.y = h[lane][pb + 1];
        hhi.x = h[lane + 16][pb + 0]; hhi.y = h[lane + 16][pb + 1];
      } else {
        hlo.x = h[lane - 16][pb + 2]; hlo.y = h[lane - 16][pb + 3];
        hhi.x = h[lane][pb + 2];      hhi.y = h[lane][pb + 3];
      }
      // 16x4 A-operand layout == 4x16 B-operand layout for f32 -> reuse regs
      c00 = __builtin_amdgcn_wmma_f32_16x16x4_f32(false, hlo, false, hlo, (short)0, c00, false, false);
      c01 = __builtin_amdgcn_wmma_f32_16x16x4_f32(false, hlo, false, hhi, (short)0, c01, false, false);
      c10 = __builtin_amdgcn_wmma_f32_16x16x4_f32(false, hhi, false, hlo, (short)0, c10, false, false);
      c11 = __builtin_amdgcn_wmma_f32_16x16x4_f32(false, hhi, false, hhi, (short)0, c11, false, false);
    }
  }

  float* __restrict__ slot = acc_ws + (blockIdx.y & (NSLOT - 1)) * WSG;

  // reduce per-channel sums
  sred[t] = spart;
  __syncthreads();
  if (t < 32) {
    float s = 0.0f;
    #pragma unroll
    for (int j = 0; j < 8; ++j) s += sred[j * 32 + t];
    atomicAdd(&slot[1024 + t], s);
  }

  // reduce Gram: D layout -> lane<16 rows 0..7, lane>=16 rows 8..15, col = lane&15
  const int rbase = (lane < 16) ? 0 : 8;
  const int col = lane & 15;
  #pragma unroll
  for (int r = 0; r < 8; ++r) {
    int row = rbase + r;
    atomicAdd(&gsh[row * 32 + col],             c00[r]);
    atomicAdd(&gsh[row * 32 + col + 16],        c01[r]);
    atomicAdd(&gsh[(row + 16) * 32 + col],      c10[r]);
    atomicAdd(&gsh[(row + 16) * 32 + col + 16], c11[r]);
  }
  __syncthreads();
  for (int i = t; i < CC * CC; i += 256) atomicAdd(&slot[i], gsh[i]);
}

// ---------------- Pass 2: fold BN1 + conv + BN2 into A (32x32) and e (32) ---
__global__ __launch_bounds__(32)
void finalize_kernel(const float* __restrict__ acc_ws,
                     const float* __restrict__ g1, const float* __restrict__ be1,
                     const float* __restrict__ w,  const float* __restrict__ g2,
                     const float* __restrict__ be2,
                     float* __restrict__ Aout, float* __restrict__ eout)
{
  __shared__ float Gsh[CC * CC];
  __shared__ float Ssh[32];
  __shared__ float mean1[32], a1s[32], d1s[32];
  const int o = threadIdx.x;

  // sum the replicated accumulator slots
  for (int i = o; i < CC * CC; i += 32) {
    float v = 0.0f;
    #pragma unroll
    for (int s = 0; s < NSLOT; ++s) v += acc_ws[s * WSG + i];
    Gsh[i] = v;
  }
  {
    float v = 0.0f;
    #pragma unroll
    for (int s = 0; s < NSLOT; ++s) v += acc_ws[s * WSG + 1024 + o];
    Ssh[o] = v;
  }
  __syncthreads();

  {
    float m = Ssh[o] / NBP;
    float var = Gsh[o * 32 + o] / NBP - m * m;
    float a = g1[o] * rsqrtf(var + EPSV);
    mean1[o] = m;
    a1s[o] = a;
    d1s[o] = be1[o] - a * m;
  }
  __syncthreads();

  float wp[32];
  float bias = 0.0f, mean2 = 0.0f;
  #pragma unroll
  for (int c = 0; c < 32; ++c) {
    float wc = w[o * 32 + c];
    wp[c] = wc * a1s[c];
    bias  += wc * d1s[c];
    mean2 += wp[c] * mean1[c];
  }
  mean2 += bias;

  // var2[o] = wp . Cov(h) . wp,  Cov = G/NBP - mean1 mean1^T
  float var2 = 0.0f;
  for (int c = 0; c < 32; ++c) {
    float acc = 0.0f;
    #pragma unroll
    for (int d = 0; d < 32; ++d) {
      float cov = Gsh[c * 32 + d] / NBP - mean1[c] * mean1[d];
      acc += cov * wp[d];
    }
    var2 += wp[c] * acc;
  }
  float a2 = g2[o] * rsqrtf(var2 + EPSV);
  float d2 = be2[o] - a2 * mean2;
  #pragma unroll
  for (int c = 0; c < 32; ++c) Aout[o * 32 + c] = a2 * wp[c];
  eout[o] = a2 * bias + d2;
}

// ---------------- Pass 3: out = leaky(A . h + e), fused WMMA GEMM -----------
__global__ __launch_bounds__(256)
void out_kernel(const float* __restrict__ bsrc, const int* __restrict__ assign,
                const float* __restrict__ Amat, const float* __restrict__ evec,
                float* __restrict__ out)
{
  __shared__ float brow[CC * NN];
  __shared__ float h[CC][HS];
  __shared__ float Ash[CC * CC];
  __shared__ float esh[CC];

  const int t = threadIdx.x;
  const int lane = t & 31;
  const int wave = t >> 5;
  const int batch = blockIdx.x;
  const int pbase = blockIdx.y * CHUNK;

  const float* bp = bsrc + batch * (CC * NN);
  for (int i = t; i < CC * NN; i += 256) brow[i] = bp[i];
  for (int i = t; i < CC * CC; i += 256) Ash[i] = Amat[i];
  if (t < 32) esh[t] = evec[t];
  __syncthreads();

  #pragma unroll 4
  for (int i = 0; i < 16; ++i) {
    int f = t + 256 * i;
    int c = f >> 7;
    int pl = f & 127;
    int p = pbase + pl;
    float v = 0.0f;
    if (p < PP) {
      int a0 = assign[2 * p];
      int a1 = assign[2 * p + 1];
      v = fmaxf(brow[c * NN + a0], brow[c * NN + a1]);
    }
    h[c][pl] = v;
  }
  __syncthreads();

  const int pl0 = wave * 16;
  const int p0 = pbase + pl0;
  if (p0 >= PP) return;   // wave-uniform tail skip (PP % 16 == 0)

  // A operands for both 16-row output tiles, 8 K-steps of 4
  v2f a0r[8], a1r[8];
  #pragma unroll
  for (int s = 0; s < 8; ++s) {
    const int k0 = 4 * s;
    if (lane < 16) {
      a0r[s].x = Ash[lane * 32 + k0];          a0r[s].y = Ash[lane * 32 + k0 + 1];
      a1r[s].x = Ash[(lane + 16) * 32 + k0];   a1r[s].y = Ash[(lane + 16) * 32 + k0 + 1];
    } else {
      a0r[s].x = Ash[(lane - 16) * 32 + k0 + 2]; a0r[s].y = Ash[(lane - 16) * 32 + k0 + 3];
      a1r[s].x = Ash[lane * 32 + k0 + 2];        a1r[s].y = Ash[lane * 32 + k0 + 3];
    }
  }

  v8f acc0 = {}, acc1 = {};
  #pragma unroll
  for (int s = 0; s < 8; ++s) {
    const int k0 = 4 * s;
    v2f bv;  // B-operand (4x16): shared by both output tiles
    if (lane < 16) {
      bv.x = h[k0 + 0][pl0 + lane];
      bv.y = h[k0 + 1][pl0 + lane];
    } else {
      bv.x = h[k0 + 2][pl0 + lane - 16];
      bv.y = h[k0 + 3][pl0 + lane - 16];
    }
    acc0 = __builtin_amdgcn_wmma_f32_16x16x4_f32(false, a0r[s], false, bv, (short)0, acc0, false, false);
    acc1 = __builtin_amdgcn_wmma_f32_16x16x4_f32(false, a1r[s], false, bv, (short)0, acc1, false, false);
  }

  // epilogue: bias + leaky relu, coalesced 16-lane row stores
  const int rbase = (lane < 16) ? 0 : 8;
  const int col = lane & 15;
  float* outp = out + (size_t)batch * CC * PP + p0 + col;
  #pragma unroll
  for (int r = 0; r < 8; ++r) {
    int row = rbase + r;
    float y0 = acc0[r] + esh[row];
    float y1 = acc1[r] + esh[row + 16];
    outp[(size_t)row * PP]        = (y0 >= 0.0f) ? y0 : NEG_SLOPE * y0;
    outp[(size_t)(row + 16) * PP] = (y1 >= 0.0f) ? y1 : NEG_SLOPE * y1;
  }
}

extern "C" void kernel_launch(void* const* d_in, const int* in_sizes, int n_in,
                              void* d_out, int out_size, void* d_ws, size_t ws_size,
                              hipStream_t stream) {
  const float* b      = (const float*)d_in[0];
  const int*   assign = (const int*)  d_in[1];
  const float* g1     = (const float*)d_in[2];
  const float* be1    = (const float*)d_in[3];
  const float* w      = (const float*)d_in[4];
  const float* g2     = (const float*)d_in[5];
  const float* be2    = (const float*)d_in[6];
  float* out = (float*)d_out;

  float* ws     = (float*)d_ws;
  float* acc_ws = ws;                    // NSLOT * WSG = 4224 floats
  float* Amat   = ws + NSLOT * WSG;      // 1024
  float* evec   = Amat + 1024;           // 32

  zero_kernel<<<(NSLOT * WSG + 255) / 256, 256, 0, stream>>>(acc_ws, NSLOT * WSG);
  stats_kernel<<<dim3(BB, SPLIT), 256, 0, stream>>>(b, assign, acc_ws);
  finalize_kernel<<<1, 32, 0, stream>>>(acc_ws, g1, be1, w, g2, be2, Amat, evec);
  out_kernel<<<dim3(BB, NCHUNK), 256, 0, stream>>>(b, assign, Amat, evec, out);
}